// Decoder_1898375544952
// MI455X (gfx1250) — compile-verified
//
#include <hip/hip_runtime.h>
#include <math.h>

// ---------------------------------------------------------------------------
// STGCN spatial stack on MI455X (gfx1250, wave32).
//  - Aggregation: f32 global atomics (L2-resident; all feature mats < 192MB L2)
//  - Dense transforms: V_WMMA_F32_16X16X4_F32 (fp32 WMMA, 16x16 tiles, K=4)
// ---------------------------------------------------------------------------

typedef float v2f __attribute__((ext_vector_type(2)));
typedef float v8f __attribute__((ext_vector_type(8)));

#define NNODES 131072
#define NEDGES 2097152

// ------------------------------ utility kernels ----------------------------

__global__ void zero_f32(float* __restrict__ p, int n) {
    int i = blockIdx.x * blockDim.x + threadIdx.x;
    if (i < n) p[i] = 0.0f;
}

__global__ void degree_kernel(const int* __restrict__ dst, float* __restrict__ deg, int E) {
    int e = blockIdx.x * blockDim.x + threadIdx.x;
    if (e < E) atomicAdd(&deg[dst[e]], 1.0f);
}

__global__ void invdeg_kernel(float* __restrict__ deg, int n) {
    int i = blockIdx.x * blockDim.x + threadIdx.x;
    if (i < n) deg[i] = 1.0f / fmaxf(deg[i], 1.0f);
}

// Scatter-add h[src] rows into agg[dst] rows. C = 1<<CLOG channels.
// Consecutive threads cover consecutive channels of one edge -> coalesced
// gather reads and coalesced atomic adds (both hit L2 on MI455X).
template <int CLOG>
__global__ void scatter_add(const float* __restrict__ h,
                            const int* __restrict__ src,
                            const int* __restrict__ dst,
                            float* __restrict__ agg, int E) {
    const int C = 1 << CLOG;
    int idx = blockIdx.x * blockDim.x + threadIdx.x;   // < E*C (max 134M, fits i32)
    if (idx >= E * C) return;
    int e = idx >> CLOG;
    int c = idx & (C - 1);
    int s = src[e];
    int d = dst[e];
    atomicAdd(&agg[d * C + c], h[s * C + c]);
}

// ------------------------------ fused WMMA GEMM ----------------------------
// OUT[16 rows, COUT] = act( X@WS + (AGG * invdeg)@WN + bias )
// One wave per 16-node tile; v_wmma_f32_16x16x4_f32 over K-chunks of 4.
//
// A (16x4 f32) fragment: lane L -> row M = L&15;
//   lanes 0-15:  v.x = K=k0+0, v.y = K=k0+1
//   lanes 16-31: v.x = K=k0+2, v.y = K=k0+3
// B (4x16 f32) fragment mirrors A (VGPR j = K-row, upper half-lanes = +2 rows).
// C/D (16x16 f32): VGPR j -> rows j (lanes 0-15) and j+8 (lanes 16-31), N = L&15.
//
// ACT: 0 = identity, 1 = relu, 2 = sigmoid.  NBR: neighbor term enabled.
template <int CIN, int COUT, int ACT, bool NBR>
__global__ __launch_bounds__(256)
void gcn_gemm(const float* __restrict__ X,
              const float* __restrict__ AGG,
              const float* __restrict__ INV,
              const float* __restrict__ WS,
              const float* __restrict__ WN,
              const float* __restrict__ BIAS,
              float* __restrict__ OUT) {
    constexpr int NT = COUT / 16;

    const int lane  = threadIdx.x & 31;
    const int wave  = threadIdx.x >> 5;
    const int tile  = blockIdx.x * 8 + wave;         // 8 waves per 256-thread block
    const int mloc  = lane & 15;
    const int row_m = (tile << 4) + mloc;            // A row this lane feeds
    const int khalf = (lane < 16) ? 0 : 2;

    v8f acc[NT];
#pragma unroll
    for (int t = 0; t < NT; ++t) acc[t] = v8f{};

    const float idg = NBR ? INV[row_m] : 0.0f;

#pragma unroll
    for (int k0 = 0; k0 < CIN; k0 += 4) {
        const int ka = k0 + khalf;

        v2f aS;
        aS.x = X[row_m * CIN + ka + 0];
        aS.y = X[row_m * CIN + ka + 1];

        v2f aN = v2f{};
        if (NBR) {
            aN.x = AGG[row_m * CIN + ka + 0] * idg;
            aN.y = AGG[row_m * CIN + ka + 1] * idg;
        }

#pragma unroll
        for (int t = 0; t < NT; ++t) {
            const int col = (t << 4) + mloc;

            v2f bS;
            bS.x = WS[(ka + 0) * COUT + col];
            bS.y = WS[(ka + 1) * COUT + col];
            acc[t] = __builtin_amdgcn_wmma_f32_16x16x4_f32(
                false, aS, false, bS, (short)0, acc[t], false, false);

            if (NBR) {
                v2f bN;
                bN.x = WN[(ka + 0) * COUT + col];
                bN.y = WN[(ka + 1) * COUT + col];
                acc[t] = __builtin_amdgcn_wmma_f32_16x16x4_f32(
                    false, aN, false, bN, (short)0, acc[t], false, false);
            }
        }
    }

    const int rbase = (tile << 4) + ((lane < 16) ? 0 : 8);
#pragma unroll
    for (int t = 0; t < NT; ++t) {
        const int col  = (t << 4) + mloc;
        const float bv = BIAS[col];
#pragma unroll
        for (int j = 0; j < 8; ++j) {
            float v = acc[t][j] + bv;
            if (ACT == 1) v = fmaxf(v, 0.0f);
            if (ACT == 2) v = 1.0f / (1.0f + __expf(-v));
            OUT[(rbase + j) * COUT + col] = v;
        }
    }
}

// ------------------------------- launcher ----------------------------------

extern "C" void kernel_launch(void* const* d_in, const int* in_sizes, int n_in,
                              void* d_out, int out_size, void* d_ws, size_t ws_size,
                              hipStream_t stream) {
    (void)in_sizes; (void)n_in; (void)out_size; (void)ws_size;

    const float* x    = (const float*)d_in[0];
    const int*   ei   = (const int*)  d_in[1];   // [2, E]
    // d_in[2] = batch (unused)
    const float* Wlin = (const float*)d_in[3];
    const float* blin = (const float*)d_in[4];
    const float* Ws3  = (const float*)d_in[5];
    const float* Wn3  = (const float*)d_in[6];
    const float* b3   = (const float*)d_in[7];
    const float* Ws2  = (const float*)d_in[8];
    const float* Wn2  = (const float*)d_in[9];
    const float* b2   = (const float*)d_in[10];
    const float* Ws1  = (const float*)d_in[11];
    const float* Wn1  = (const float*)d_in[12];
    const float* b1   = (const float*)d_in[13];
    float*       out  = (float*)d_out;

    const int N = NNODES;
    const int E = NEDGES;
    const int* src = ei;
    const int* dst = ei + E;

    // Workspace layout (floats): h0[N*16] h1[N*32] h2[N*64] agg[N*64] inv[N]
    float* h0  = (float*)d_ws;
    float* h1  = h0  + (size_t)N * 16;
    float* h2  = h1  + (size_t)N * 32;
    float* agg = h2  + (size_t)N * 64;
    float* inv = agg + (size_t)N * 64;

    const int TPB = 256;
    const int gemm_blocks = N / 128;   // 8192 tiles / 8 waves per block

    // ---- degree -> invdeg (shared by all three blocks) ----
    zero_f32<<<(N + TPB - 1) / TPB, TPB, 0, stream>>>(inv, N);
    degree_kernel<<<(E + TPB - 1) / TPB, TPB, 0, stream>>>(dst, inv, E);
    invdeg_kernel<<<(N + TPB - 1) / TPB, TPB, 0, stream>>>(inv, N);

    // ---- h0 = x @ W_lin + b_lin  (16 -> 16, no neighbor term, identity) ----
    gcn_gemm<16, 16, 0, false><<<gemm_blocks, TPB, 0, stream>>>(
        x, nullptr, nullptr, Wlin, nullptr, blin, h0);

    // ---- block3: 16 -> 32, relu ----
    zero_f32<<<(N * 16 + TPB - 1) / TPB, TPB, 0, stream>>>(agg, N * 16);
    scatter_add<4><<<(E * 16 + TPB - 1) / TPB, TPB, 0, stream>>>(h0, src, dst, agg, E);
    gcn_gemm<16, 32, 1, true><<<gemm_blocks, TPB, 0, stream>>>(
        h0, agg, inv, Ws3, Wn3, b3, h1);

    // ---- block2: 32 -> 64, relu ----
    zero_f32<<<(N * 32 + TPB - 1) / TPB, TPB, 0, stream>>>(agg, N * 32);
    scatter_add<5><<<(E * 32 + TPB - 1) / TPB, TPB, 0, stream>>>(h1, src, dst, agg, E);
    gcn_gemm<32, 64, 1, true><<<gemm_blocks, TPB, 0, stream>>>(
        h1, agg, inv, Ws2, Wn2, b2, h2);

    // ---- block1: 64 -> 128, sigmoid -> d_out ----
    zero_f32<<<(N * 64 + TPB - 1) / TPB, TPB, 0, stream>>>(agg, N * 64);
    scatter_add<6><<<(E * 64 + TPB - 1) / TPB, TPB, 0, stream>>>(h2, src, dst, agg, E);
    gcn_gemm<64, 128, 2, true><<<gemm_blocks, TPB, 0, stream>>>(
        h2, agg, inv, Ws1, Wn1, b1, out);
}